// FPS_69595650064384
// MI455X (gfx1250) — compile-verified
//
#include <hip/hip_runtime.h>

// FPS: B=4 batches, N=8192 points, S=1024 samples.
// One workgroup per batch; points cached in LDS (96KB via TDM DMA),
// per-thread squared distances in VGPRs; one barrier per round.

#define FPS_N   8192
#define FPS_S   1024
#define FPS_B   4
#define FPS_T   512
#define FPS_PPT (FPS_N / FPS_T)   // 16 points per thread
#define FPS_NW  (FPS_T / 32)      // 16 waves

typedef unsigned int tdm_v4u __attribute__((ext_vector_type(4)));
typedef int          tdm_v8i __attribute__((ext_vector_type(8)));
typedef int          tdm_v4i __attribute__((ext_vector_type(4)));

#ifndef __has_builtin
#define __has_builtin(x) 0
#endif

#if __has_builtin(__builtin_amdgcn_tensor_load_to_lds)
#define FPS_HAVE_TDM 1
#else
#define FPS_HAVE_TDM 0
#endif

__device__ __forceinline__ unsigned long long fps_shfl_xor_u64(unsigned long long v, int m) {
  unsigned lo = __shfl_xor((unsigned)v, m, 32);
  unsigned hi = __shfl_xor((unsigned)(v >> 32), m, 32);
  return ((unsigned long long)hi << 32) | (unsigned long long)lo;
}

__global__ void __launch_bounds__(FPS_T)
fps_kernel(const float* __restrict__ in, float* __restrict__ out) {
  const int b    = blockIdx.x;
  const int t    = threadIdx.x;
  const int lane = t & 31;
  const int wid  = t >> 5;

  extern __shared__ float lds_pts[];                     // FPS_N*3 floats = 96KB
  __shared__ unsigned long long partials[2][FPS_NW];     // double-buffered wave maxima

  const float* bp = in  + (size_t)b * (FPS_N * 3);
  float*       op = out + (size_t)b * (FPS_S * 3);

#if FPS_HAVE_TDM
  // --- Tensor Data Mover: bulk DMA the whole 96KB point tile into LDS -------
  if (wid == 0) {
    unsigned long long ga    = (unsigned long long)(uintptr_t)bp;   // global tile addr
    unsigned           lbase = (unsigned)(uintptr_t)&lds_pts[0];    // LDS byte offset (low 32 bits of flat LDS addr)

    // D# group 0: count=1, lds_addr, global_addr[56:0], type=2 ("image")
    tdm_v4u g0;
    g0[0] = 1u;
    g0[1] = lbase;
    g0[2] = (unsigned)(ga & 0xFFFFFFFFull);
    g0[3] = ((unsigned)((ga >> 32) & 0x1FFFFFFull)) | (2u << 30);

    // D# group 1: mask=0 (not in cluster), data_size=4B (code 2),
    // tensor_dim0 = tile_dim0 = 24576 elems, tensor_dim1 = tile_dim1 = 1 row,
    // tensor_dim0_stride = 24576. (1-row 2D tile == flat 96KB copy.)
    const unsigned long long DIM0 = (unsigned long long)(FPS_N * 3); // 24576 (0x6000)
    unsigned long long q0 = (2ull << 16) | ((DIM0 & 0xFFFFull) << 48);
    unsigned long long q1 = ((DIM0 >> 16) & 0xFFFFull)               // tensor_dim0[31:16]
                          | (1ull << 16)                             // tensor_dim1 = 1
                          | ((DIM0 & 0xFFFFull) << 48);              // tile_dim0
    unsigned long long q2 = 1ull                                     // tile_dim1 = 1
                          | ((DIM0 & 0xFFFFFFFFull) << 32);          // tensor_dim0_stride[31:0]
    unsigned long long q3 = 0ull;                                    // stride0 hi / stride1 = 0
    tdm_v8i g1;
    g1[0] = (int)(unsigned)q0; g1[1] = (int)(unsigned)(q0 >> 32);
    g1[2] = (int)(unsigned)q1; g1[3] = (int)(unsigned)(q1 >> 32);
    g1[4] = (int)(unsigned)q2; g1[5] = (int)(unsigned)(q2 >> 32);
    g1[6] = (int)(unsigned)q3; g1[7] = (int)(unsigned)(q3 >> 32);
    tdm_v4i g2 = {0, 0, 0, 0};
    tdm_v4i g3 = {0, 0, 0, 0};
#if defined(__clang_major__) && (__clang_major__ >= 23)
    tdm_v8i g4z = {0, 0, 0, 0, 0, 0, 0, 0};   // amdgpu-toolchain 6-arg form
    __builtin_amdgcn_tensor_load_to_lds(g0, g1, g2, g3, g4z, 0);
#else
    __builtin_amdgcn_tensor_load_to_lds(g0, g1, g2, g3, 0);          // ROCm 7.2 5-arg form
#endif
  }
#endif

  // --- each thread loads its own 16 points into registers (overlaps the DMA)
  float px[FPS_PPT], py[FPS_PPT], pz[FPS_PPT], ds2[FPS_PPT];
  {
    float a[FPS_PPT * 3];
    const float4* vp = (const float4*)(bp + t * (FPS_PPT * 3));      // 192B per thread, 16B aligned
#pragma unroll
    for (int q = 0; q < (FPS_PPT * 3) / 4; ++q) {
      float4 v = vp[q];
      a[4 * q + 0] = v.x; a[4 * q + 1] = v.y; a[4 * q + 2] = v.z; a[4 * q + 3] = v.w;
#if !FPS_HAVE_TDM
      ((float4*)lds_pts)[t * ((FPS_PPT * 3) / 4) + q] = v;           // fallback LDS fill
#endif
    }
#pragma unroll
    for (int k = 0; k < FPS_PPT; ++k) {
      px[k] = a[3 * k + 0]; py[k] = a[3 * k + 1]; pz[k] = a[3 * k + 2];
    }
  }

#if FPS_HAVE_TDM
  if (wid == 0) {
#if __has_builtin(__builtin_amdgcn_s_wait_tensorcnt)
    __builtin_amdgcn_s_wait_tensorcnt(0);
#else
    asm volatile("s_wait_tensorcnt 0x0" ::: "memory");
#endif
  }
#endif
  __syncthreads();

  // --- ds initialized to squared distance from point 0 ----------------------
  {
    float x0 = lds_pts[0], y0 = lds_pts[1], z0 = lds_pts[2];
#pragma unroll
    for (int k = 0; k < FPS_PPT; ++k) {
      float dx = px[k] - x0, dy = py[k] - y0, dz = pz[k] - z0;
      ds2[k] = fmaf(dx, dx, fmaf(dy, dy, dz * dz));
    }
  }

  const int base = t * FPS_PPT;

  for (int s = 0; s < FPS_S; ++s) {
    // local argmax over 16 register values; ascending scan with strict '>'
    // keeps the lowest index on ties (matches jnp.argmax)
    float bv = ds2[0]; int bi = 0;
#pragma unroll
    for (int k = 1; k < FPS_PPT; ++k)
      if (ds2[k] > bv) { bv = ds2[k]; bi = k; }

    // key: d2 >= 0 so IEEE bits are order-isomorphic; low 13 bits = 8191-idx
    // => u64 max == (max d2, tie -> min idx)
    unsigned long long key =
        ((unsigned long long)__float_as_uint(bv) << 13) |
        (unsigned long long)(unsigned)(8191 - (base + bi));

    // wave32 butterfly max
#pragma unroll
    for (int m = 1; m < 32; m <<= 1) {
      unsigned long long o = fps_shfl_xor_u64(key, m);
      if (o > key) key = o;
    }

    const int buf = s & 1;                 // double-buffered partials: 1 barrier/round
    if (lane == 0) partials[buf][wid] = key;
    __syncthreads();

    // every wave redundantly reduces the 16 partials -> all threads know winner
    unsigned long long k2 = partials[buf][lane & (FPS_NW - 1)];
#pragma unroll
    for (int m = 1; m < 32; m <<= 1) {
      unsigned long long o = fps_shfl_xor_u64(k2, m);
      if (o > k2) k2 = o;
    }

    int   widx = 8191 - (int)(k2 & 8191ull);
    float sx = lds_pts[widx * 3 + 0];      // uniform addr -> LDS broadcast
    float sy = lds_pts[widx * 3 + 1];
    float sz = lds_pts[widx * 3 + 2];

    if (t == 0) {
      op[s * 3 + 0] = sx; op[s * 3 + 1] = sy; op[s * 3 + 2] = sz;
    }

    // min-update of squared distances (all in registers)
#pragma unroll
    for (int k = 0; k < FPS_PPT; ++k) {
      float dx = px[k] - sx, dy = py[k] - sy, dz = pz[k] - sz;
      float d2 = fmaf(dx, dx, fmaf(dy, dy, dz * dz));
      ds2[k] = fminf(ds2[k], d2);
    }
  }
}

extern "C" void kernel_launch(void* const* d_in, const int* in_sizes, int n_in,
                              void* d_out, int out_size, void* d_ws, size_t ws_size,
                              hipStream_t stream) {
  (void)in_sizes; (void)n_in; (void)out_size; (void)d_ws; (void)ws_size;
  const float* in  = (const float*)d_in[0];
  float*       out = (float*)d_out;
  size_t shmem = (size_t)FPS_N * 3 * sizeof(float);   // 96KB dynamic LDS (WGP has 320KB)
  fps_kernel<<<dim3(FPS_B), dim3(FPS_T), shmem, stream>>>(in, out);
}